// GraphSynMasker_33655363731849
// MI455X (gfx1250) — compile-verified
//
#include <hip/hip_runtime.h>
#include <math.h>

#define NN 50000
#define NE 800000
#define DIM 128
#define NGRAPH 512

typedef float v2f __attribute__((ext_vector_type(2)));
typedef float v8f __attribute__((ext_vector_type(8)));

__device__ __forceinline__ void atomAddF(float* p, float v) {
    unsafeAtomicAdd(p, v);   // global_atomic_add_f32
}

// ---------------- degree counting (out-deg for norm, in-deg for CSR-T) ----------------
__global__ void zero_u32(unsigned* p, int n) {
    int i = blockIdx.x * blockDim.x + threadIdx.x;
    if (i < n) p[i] = 0u;
}
__global__ void zero_f32(float* p, int n) {
    int i = blockIdx.x * blockDim.x + threadIdx.x;
    if (i < n) p[i] = 0.0f;
}
__global__ void copy_u32(const unsigned* __restrict__ s, unsigned* __restrict__ d, int n) {
    int i = blockIdx.x * blockDim.x + threadIdx.x;
    if (i < n) d[i] = s[i];
}
__global__ void count_deg(const int* __restrict__ row, const int* __restrict__ col,
                          unsigned* __restrict__ cntR, unsigned* __restrict__ cntC, int E) {
    int e = blockIdx.x * blockDim.x + threadIdx.x;
    if (e < E) {
        atomicAdd(&cntR[row[e]], 1u);
        atomicAdd(&cntC[col[e]], 1u);
    }
}
__global__ void dis_kernel(const unsigned* __restrict__ cnt, float* __restrict__ dis, int N) {
    int v = blockIdx.x * blockDim.x + threadIdx.x;
    if (v < N) dis[v] = rsqrtf((float)(cnt[v] + 1u));   // +1 self loop
}

// ---------------- prefix sum (3-phase) for CSR-T offsets ----------------
__global__ __launch_bounds__(256) void block_reduce(const unsigned* __restrict__ cnt,
                                                    unsigned* __restrict__ bsum, int n) {
    __shared__ unsigned s[256];
    int i = blockIdx.x * 256 + threadIdx.x;
    s[threadIdx.x] = (i < n) ? cnt[i] : 0u;
    __syncthreads();
    for (int off = 128; off > 0; off >>= 1) {
        if (threadIdx.x < off) s[threadIdx.x] += s[threadIdx.x + off];
        __syncthreads();
    }
    if (threadIdx.x == 0) bsum[blockIdx.x] = s[0];
}
__global__ void scan_bsums(unsigned* bsum, int nb) {   // tiny serial scan
    if (threadIdx.x == 0 && blockIdx.x == 0) {
        unsigned acc = 0;
        for (int i = 0; i < nb; ++i) { unsigned t = bsum[i]; bsum[i] = acc; acc += t; }
    }
}
__global__ __launch_bounds__(256) void block_scan(const unsigned* __restrict__ cnt,
                                                  const unsigned* __restrict__ bsum,
                                                  unsigned* __restrict__ offs, int n) {
    __shared__ unsigned s[256];
    int i = blockIdx.x * 256 + threadIdx.x;
    unsigned v = (i < n) ? cnt[i] : 0u;
    s[threadIdx.x] = v;
    __syncthreads();
    for (int off = 1; off < 256; off <<= 1) {    // Hillis-Steele inclusive scan
        unsigned t = (threadIdx.x >= off) ? s[threadIdx.x - off] : 0u;
        __syncthreads();
        s[threadIdx.x] += t;
        __syncthreads();
    }
    unsigned incl = bsum[blockIdx.x] + s[threadIdx.x];
    if (i < n) offs[i] = incl - v;   // exclusive
    if (i == n - 1) offs[n] = incl;  // total == E
}
__global__ void fill_csr(const int* __restrict__ row, const int* __restrict__ col,
                         unsigned* __restrict__ pos, int* __restrict__ srcs, int E) {
    int e = blockIdx.x * blockDim.x + threadIdx.x;
    if (e < E) {
        unsigned slot = atomicAdd(&pos[col[e]], 1u);
        srcs[slot] = row[e];
    }
}

// ---------------- fold BN into weights ----------------
__global__ __launch_bounds__(128) void prep_weff(
    const float* __restrict__ W, const float* __restrict__ gamma,
    const float* __restrict__ beta, const float* __restrict__ mean,
    const float* __restrict__ var, float* __restrict__ Weff, float* __restrict__ beff)
{
    __shared__ float red[DIM];
    int j = blockIdx.x;
    int k = threadIdx.x;
    float s = gamma[k] * rsqrtf(var[k] + 1e-5f);
    float w = W[k * DIM + j];
    Weff[k * DIM + j] = s * w;
    red[k] = (beta[k] - mean[k] * s) * w;
    __syncthreads();
    for (int off = 64; off > 0; off >>= 1) {
        if (k < off) red[k] += red[k + off];
        __syncthreads();
    }
    if (k == 0) beff[j] = red[0];
}

// ---------------- WMMA GEMM: out = act(act_in(A)[M,128] @ Weff + bias) ----------------
// LDS holds Weff in K-pair-interleaved float2 so each B frag is one aligned ds_load_b64.
// A frag software-pipelined 1 deep. 8 waves/block, 16 rows/wave; M % 16 == 0.
__global__ __launch_bounds__(256) void gemm_wmma(
    const float* __restrict__ A, const float* __restrict__ W,
    const float* __restrict__ bias, float* __restrict__ out, int M,
    int relu_in, int relu_out)
{
    __shared__ float sWf[DIM * DIM];  // 64 KB, pair layout: [(k>>1)*128 + n]*2 + (k&1)
    {
        const float4* gw4 = (const float4*)W;
        for (int i = threadIdx.x; i < DIM * DIM / 4; i += 256) {
            float4 w = gw4[i];
            int k = i >> 5, n0 = (i & 31) << 2;
            int e = (k & 1);
            int b = ((k >> 1) * DIM + n0) * 2 + e;
            sWf[b + 0] = w.x;
            sWf[b + 2] = w.y;
            sWf[b + 4] = w.z;
            sWf[b + 6] = w.w;
        }
    }
    __syncthreads();
    const v2f* sW2 = (const v2f*)sWf;

    const int wave = threadIdx.x >> 5;
    const int lane = threadIdx.x & 31;
    const int half = lane >> 4;     // lanes 0-15: K pair {0,1}; lanes 16-31: K pair {2,3}
    const int lm   = lane & 15;
    const int m0 = blockIdx.x * 128 + wave * 16;
    if (m0 >= M) return;            // wave-uniform -> EXEC all ones for WMMA

    v8f acc[8] = {};
    const float* __restrict__ ap = A + (size_t)(m0 + lm) * DIM + 2 * half;

    v2f a = *(const v2f*)ap;
    if (relu_in) { a.x = fmaxf(a.x, 0.f); a.y = fmaxf(a.y, 0.f); }
    for (int kk = 0; kk < DIM; kk += 4) {
        const int kn = (kk + 4 < DIM) ? (kk + 4) : kk;   // clamp: redundant last reload
        v2f an = *(const v2f*)(ap + kn);
        if (relu_in) { an.x = fmaxf(an.x, 0.f); an.y = fmaxf(an.y, 0.f); }
        const int bb = ((kk >> 1) + half) * DIM + lm;
        #pragma unroll
        for (int nt = 0; nt < 8; ++nt) {
            v2f b = sW2[bb + nt * 16];
            acc[nt] = __builtin_amdgcn_wmma_f32_16x16x4_f32(
                false, a, false, b, (short)0, acc[nt], false, false);
        }
        a = an;
    }

    #pragma unroll
    for (int nt = 0; nt < 8; ++nt) {
        const int n = nt * 16 + lm;
        const float bv = bias[n];
        #pragma unroll
        for (int r = 0; r < 8; ++r) {
            const int m = m0 + r + 8 * half;
            float v = acc[nt][r] + bv;
            if (relu_out) v = fmaxf(v, 0.0f);
            out[(size_t)m * DIM + n] = v;
        }
    }
}

// ---------------- GCN propagation by gather (CSR-T), fused self-loop + bias ----------------
// wave per destination node; 4 features/lane; no write atomics.
__global__ __launch_bounds__(256) void gather_nodes(
    const float* __restrict__ t, const int* __restrict__ srcs,
    const unsigned* __restrict__ offs, const float* __restrict__ dis,
    const float* __restrict__ bh, float* __restrict__ h, int N)
{
    int c = blockIdx.x * 8 + (threadIdx.x >> 5);
    int lane = threadIdx.x & 31;
    if (c >= N) return;
    float dc = dis[c];
    float4 tv = ((const float4*)t)[(size_t)c * 32 + lane];
    float4 bv = ((const float4*)bh)[lane];
    float d2 = dc * dc;
    float4 acc = { d2 * tv.x + bv.x, d2 * tv.y + bv.y, d2 * tv.z + bv.z, d2 * tv.w + bv.w };
    unsigned s0 = offs[c], s1 = offs[c + 1];
    for (unsigned i = s0; i < s1; ++i) {
        int r = srcs[i];                 // broadcast load
        float w = dis[r] * dc;           // broadcast load
        float4 rv = ((const float4*)t)[(size_t)r * 32 + lane];  // coalesced 512B/wave
        acc.x += w * rv.x; acc.y += w * rv.y; acc.z += w * rv.z; acc.w += w * rv.w;
    }
    ((float4*)h)[(size_t)c * 32 + lane] = acc;
}

// ---------------- heads + fused per-graph stats ----------------
__global__ __launch_bounds__(256) void node_heads_stats(
    const float* __restrict__ h, const float* __restrict__ Wn,
    const float* __restrict__ We, const float* __restrict__ bnb,
    const int* __restrict__ batch,
    float* __restrict__ nodekey, float* __restrict__ pr, float* __restrict__ pc,
    float* __restrict__ acc, int N)
{
    int v = blockIdx.x * 8 + (threadIdx.x >> 5);
    int lane = threadIdx.x & 31;
    if (v >= N) return;
    float4 hv = ((const float4*)h)[(size_t)v * 32 + lane];
    float4 wn = ((const float4*)Wn)[lane];
    float4 wr = ((const float4*)We)[lane];
    float4 wc = ((const float4*)We)[32 + lane];
    float dn = hv.x * wn.x + hv.y * wn.y + hv.z * wn.z + hv.w * wn.w;
    float dr = hv.x * wr.x + hv.y * wr.y + hv.z * wr.z + hv.w * wr.w;
    float dc = hv.x * wc.x + hv.y * wc.y + hv.z * wc.z + hv.w * wc.w;
    for (int off = 16; off > 0; off >>= 1) {
        dn += __shfl_down(dn, off, 32);
        dr += __shfl_down(dr, off, 32);
        dc += __shfl_down(dc, off, 32);
    }
    if (lane == 0) {
        float k = 1.0f / (1.0f + expf(-(dn + bnb[0])));
        nodekey[v] = k;
        pr[v] = dr;
        pc[v] = dc;
        int g = batch[v];
        atomAddF(acc + g, k);
        atomAddF(acc + 512 + g, 1.0f - k);
        atomAddF(acc + 1024 + g, k > 0.0f ? 1.0f : 0.0f);
        atomAddF(acc + 1536 + g, 1.0f);
    }
}

__global__ void edge_heads_stats(const int* __restrict__ row, const int* __restrict__ col,
                                 const float* __restrict__ pr, const float* __restrict__ pc,
                                 const float* __restrict__ be, const int* __restrict__ batch,
                                 float* __restrict__ ekey, float* __restrict__ acc, int E)
{
    int e = blockIdx.x * blockDim.x + threadIdx.x;
    if (e >= E) return;
    int r = row[e];
    float s = pr[r] + pc[col[e]] + be[0];
    float k = 1.0f / (1.0f + expf(-s));
    ekey[e] = k;
    int g = batch[r];
    atomAddF(acc + g, k);
    atomAddF(acc + 512 + g, 1.0f - k);
    atomAddF(acc + 1024 + g, k > 0.0f ? 1.0f : 0.0f);
    atomAddF(acc + 1536 + g, 1.0f);
}

// out order: node_key_num, node_env_num, edge_key_num, edge_env_num, nz_node, nz_edge
__global__ void finalize_stats(const float* __restrict__ aN, const float* __restrict__ aE,
                               float* __restrict__ o)
{
    int g = blockIdx.x * blockDim.x + threadIdx.x;
    if (g >= NGRAPH) return;
    o[g]        = aN[g] + 1e-8f;
    o[512 + g]  = aN[512 + g] + 1e-8f;
    o[1024 + g] = aE[g] + 1e-8f;
    o[1536 + g] = aE[512 + g] + 1e-8f;
    o[2048 + g] = aN[1024 + g] / aN[1536 + g];
    o[2560 + g] = aE[1024 + g] / aE[1536 + g];
}

extern "C" void kernel_launch(void* const* d_in, const int* in_sizes, int n_in,
                              void* d_out, int out_size, void* d_ws, size_t ws_size,
                              hipStream_t stream)
{
    const float* x        = (const float*)d_in[0];
    const int*   ei       = (const int*)d_in[1];
    const int*   batch    = (const int*)d_in[2];
    const float* W        = (const float*)d_in[4];
    const float* b_hidden = (const float*)d_in[5];
    const float* gam      = (const float*)d_in[6];
    const float* bet      = (const float*)d_in[7];
    const float* mu       = (const float*)d_in[8];
    const float* var      = (const float*)d_in[9];
    const float* We       = (const float*)d_in[10];
    const float* be       = (const float*)d_in[11];
    const float* Wn       = (const float*)d_in[12];
    const float* bnb      = (const float*)d_in[13];

    const int N = NN, E = NE;
    const int* row = ei;
    const int* col = ei + E;

    char* ws = (char*)d_ws;
    float*    bufA = (float*)(ws);                 // 25,600,000 B
    float*    bufB = (float*)(ws + 25600000);      // 25,600,000 B
    float*    dis  = (float*)(ws + 51200000);      //    200,000 B
    unsigned* cntR = (unsigned*)(ws + 51400000);   //    200,000 B
    unsigned* cntC = (unsigned*)(ws + 51600000);   //    200,000 B  (contiguous w/ cntR)
    unsigned* offs = (unsigned*)(ws + 51800000);   //    200,016 B  (N+1 entries)
    unsigned* pos  = (unsigned*)(ws + 52000016);   //    200,000 B
    int*      srcs = (int*)     (ws + 52200016);   //  3,200,000 B
    unsigned* bsum = (unsigned*)(ws + 55400016);   //      1,024 B
    float*    Weff = (float*)   (ws + 55401040);   //     65,536 B
    float*    beff = (float*)   (ws + 55466576);   //        512 B
    float*    pr   = (float*)   (ws + 55467088);   //    200,000 B
    float*    pc   = (float*)   (ws + 55667088);   //    200,000 B
    float*    accN = (float*)   (ws + 55867088);   //      8,192 B
    float*    accE = (float*)   (ws + 55875280);   //      8,192 B (contiguous w/ accN)

    float* o_nodekey = (float*)d_out;
    float* o_edgekey = o_nodekey + N;
    float* o_stats   = o_edgekey + E;

    const int nb = (N + 255) / 256;   // 196 scan blocks

    // degrees + symmetric norm + transposed CSR (built once, used twice)
    zero_u32<<<(2 * N + 255) / 256, 256, 0, stream>>>(cntR, 2 * N);
    count_deg<<<(E + 255) / 256, 256, 0, stream>>>(row, col, cntR, cntC, E);
    dis_kernel<<<(N + 255) / 256, 256, 0, stream>>>(cntR, dis, N);
    block_reduce<<<nb, 256, 0, stream>>>(cntC, bsum, N);
    scan_bsums<<<1, 32, 0, stream>>>(bsum, nb);
    block_scan<<<nb, 256, 0, stream>>>(cntC, bsum, offs, N);
    copy_u32<<<(N + 255) / 256, 256, 0, stream>>>(offs, pos, N);
    fill_csr<<<(E + 255) / 256, 256, 0, stream>>>(row, col, pos, srcs, E);

    const int gemm_blocks = (N + 127) / 128;

    // layer 1: h = relu(BN0(x) @ W0)
    prep_weff<<<128, 128, 0, stream>>>(W, gam, bet, mu, var, Weff, beff);
    gemm_wmma<<<gemm_blocks, 256, 0, stream>>>(x, Weff, beff, bufA, N, 0, 1);

    // layer 2: h = relu(prop(BN1(h) @ W1) + bh0)  (relu folded into next GEMM's A-load)
    prep_weff<<<128, 128, 0, stream>>>(W + DIM * DIM, gam + DIM, bet + DIM, mu + DIM, var + DIM, Weff, beff);
    gemm_wmma<<<gemm_blocks, 256, 0, stream>>>(bufA, Weff, beff, bufB, N, 0, 0);
    gather_nodes<<<(N + 7) / 8, 256, 0, stream>>>(bufB, srcs, offs, dis, b_hidden, bufA, N);

    // layer 3: node_rep = prop(BN2(relu(h)) @ W2) + bh1
    prep_weff<<<128, 128, 0, stream>>>(W + 2 * DIM * DIM, gam + 2 * DIM, bet + 2 * DIM, mu + 2 * DIM, var + 2 * DIM, Weff, beff);
    gemm_wmma<<<gemm_blocks, 256, 0, stream>>>(bufA, Weff, beff, bufB, N, 1, 0);
    gather_nodes<<<(N + 7) / 8, 256, 0, stream>>>(bufB, srcs, offs, dis, b_hidden + DIM, bufA, N);

    // heads + per-graph stats
    zero_f32<<<(4096 + 255) / 256, 256, 0, stream>>>(accN, 4096);  // accN+accE contiguous
    node_heads_stats<<<(N + 7) / 8, 256, 0, stream>>>(bufA, Wn, We, bnb, batch,
                                                      o_nodekey, pr, pc, accN, N);
    edge_heads_stats<<<(E + 255) / 256, 256, 0, stream>>>(row, col, pr, pc, be, batch,
                                                          o_edgekey, accE, E);
    finalize_stats<<<(NGRAPH + 255) / 256, 256, 0, stream>>>(accN, accE, o_stats);
}